// mGRU_33526514712651
// MI455X (gfx1250) — compile-verified
//
#include <hip/hip_runtime.h>

typedef float v2f __attribute__((ext_vector_type(2)));
typedef float v8f __attribute__((ext_vector_type(8)));

#define H 4096
#define WAVES_PER_BLOCK 16
#define BLOCK (WAVES_PER_BLOCK * 32)
#define KCHUNK (H / WAVES_PER_BLOCK)  // 256 K-elements per wave

enum { MODE_XTILDE = 0, MODE_SIGMOID = 1, MODE_SIGMOID_MUL = 2, MODE_TANH = 3 };

// out_pre[n] = sum_k a[k]*Wa[k][n] + sum_k b[k]*Wb[k][n] + bias[n], then per-MODE epilogue.
// One workgroup (16 waves) per 16-column output tile; each wave covers a K-chunk of 256
// for BOTH matrices via V_WMMA_F32_16X16X4_F32 with the input vector replicated across
// all 16 A-rows (D rows are then identical; row 0 = the 16 partial outputs).
template <int MODE>
__global__ __launch_bounds__(BLOCK) void gemv2_wmma(
    const float* __restrict__ a, const float* __restrict__ Wa,
    const float* __restrict__ b, const float* __restrict__ Wb,
    const float* __restrict__ bias, const float* __restrict__ extra,
    float* __restrict__ out) {
  __shared__ float red[WAVES_PER_BLOCK * 16];

  const int tid  = threadIdx.x;
  const int wave = tid >> 5;
  const int lane = tid & 31;
  const int half = lane >> 4;   // 0: lanes 0-15 (K0/K1), 1: lanes 16-31 (K2/K3)
  const int nl   = lane & 15;   // column within the 16-wide tile
  const int nb   = blockIdx.x * 16;

  // Column base pointers for the B-matrix (4x16 f32) tiles.
  const float* __restrict__ wa_col = Wa + nb + nl;
  const float* __restrict__ wb_col = Wb + nb + nl;

  v8f acc0 = {};  // accumulates a @ Wa
  v8f acc1 = {};  // accumulates b @ Wb

  // Per-lane K index for this lane's A/B register slots (K = kk, kk+1).
  int kk = wave * KCHUNK + 2 * half;

#pragma unroll 2
  for (int it = 0; it < KCHUNK / 4; ++it, kk += 4) {
    // A (16x4 f32): all 16 rows replicated with the vector chunk.
    // VGPR0 = vec[kk], VGPR1 = vec[kk+1] (lanes 0-15 -> K0/K1, lanes 16-31 -> K2/K3).
    v2f av = *(const v2f*)(a + kk);
    v2f bv = *(const v2f*)(b + kk);

    // B (4x16 f32): VGPR0 lane nl = W[kk][nb+nl], VGPR1 = W[kk+1][nb+nl].
    v2f wav, wbv;
    wav.x = wa_col[kk * H];
    wav.y = wa_col[(kk + 1) * H];
    wbv.x = wb_col[kk * H];
    wbv.y = wb_col[(kk + 1) * H];

    acc0 = __builtin_amdgcn_wmma_f32_16x16x4_f32(false, av, false, wav,
                                                 (short)0, acc0, false, false);
    acc1 = __builtin_amdgcn_wmma_f32_16x16x4_f32(false, bv, false, wbv,
                                                 (short)0, acc1, false, false);
  }

  // D rows are replicated; row 0 lives in VGPR0 (element 0) of lanes 0-15.
  if (lane < 16) red[wave * 16 + lane] = acc0[0] + acc1[0];
  __syncthreads();

  if (tid < 16) {
    float s = 0.0f;
#pragma unroll
    for (int w = 0; w < WAVES_PER_BLOCK; ++w) s += red[w * 16 + tid];
    s += bias[nb + tid];

    float r;
    if (MODE == MODE_XTILDE) {
      r = extra[nb + tid] * s;                       // x_tilde = x_t * m_t
    } else if (MODE == MODE_SIGMOID) {
      r = 1.0f / (1.0f + expf(-s));                  // z_t
    } else if (MODE == MODE_SIGMOID_MUL) {
      r = (1.0f / (1.0f + expf(-s))) * extra[nb + tid];  // r_t * h_prev
    } else {
      r = tanhf(s);                                  // h_tilde
    }
    out[nb + tid] = r;
  }
}

__global__ void combine_kernel(const float* __restrict__ z,
                               const float* __restrict__ h_prev,
                               const float* __restrict__ h_tilde,
                               float* __restrict__ out) {
  int i = blockIdx.x * blockDim.x + threadIdx.x;
  if (i < H) {
    float zi = z[i];
    out[i] = (1.0f - zi) * h_prev[i] + zi * h_tilde[i];
  }
}

extern "C" void kernel_launch(void* const* d_in, const int* in_sizes, int n_in,
                              void* d_out, int out_size, void* d_ws, size_t ws_size,
                              hipStream_t stream) {
  const float* x_t    = (const float*)d_in[0];
  const float* h_prev = (const float*)d_in[1];
  const float* W_m    = (const float*)d_in[2];
  const float* W_z    = (const float*)d_in[3];
  const float* W_r    = (const float*)d_in[4];
  const float* W_h    = (const float*)d_in[5];
  const float* U_m    = (const float*)d_in[6];
  const float* U_z    = (const float*)d_in[7];
  const float* U_r    = (const float*)d_in[8];
  const float* U_h    = (const float*)d_in[9];
  const float* b_m    = (const float*)d_in[10];
  const float* b_z    = (const float*)d_in[11];
  const float* b_r    = (const float*)d_in[12];
  const float* b_h    = (const float*)d_in[13];

  float* ws      = (float*)d_ws;
  float* x_tilde = ws + 0 * H;
  float* z_t     = ws + 1 * H;
  float* rh      = ws + 2 * H;
  float* h_tilde = ws + 3 * H;
  float* out     = (float*)d_out;

  dim3 grid(H / 16);
  dim3 block(BLOCK);

  // Phase 1: m_t = x@W_m + h@U_m + b_m; x_tilde = x * m_t
  gemv2_wmma<MODE_XTILDE><<<grid, block, 0, stream>>>(
      x_t, W_m, h_prev, U_m, b_m, x_t, x_tilde);
  // Phase 2a: z_t = sigmoid(x_tilde@W_z + h@U_z + b_z)
  gemv2_wmma<MODE_SIGMOID><<<grid, block, 0, stream>>>(
      x_tilde, W_z, h_prev, U_z, b_z, x_t /*unused*/, z_t);
  // Phase 2b: rh = sigmoid(x_tilde@W_r + h@U_r + b_r) * h_prev
  gemv2_wmma<MODE_SIGMOID_MUL><<<grid, block, 0, stream>>>(
      x_tilde, W_r, h_prev, U_r, b_r, h_prev, rh);
  // Phase 3: h_tilde = tanh(x_tilde@W_h + rh@U_h + b_h)
  gemv2_wmma<MODE_TANH><<<grid, block, 0, stream>>>(
      x_tilde, W_h, rh, U_h, b_h, x_t /*unused*/, h_tilde);
  // Phase 4: h_t = (1-z)*h_prev + z*h_tilde
  combine_kernel<<<(H + 255) / 256, 256, 0, stream>>>(z_t, h_prev, h_tilde, out);
}